// ThalamusRouter_26551487824577
// MI455X (gfx1250) — compile-verified
//
#include <hip/hip_runtime.h>
#include <hip/hip_bf16.h>

typedef float v2f __attribute__((ext_vector_type(2)));
typedef float v8f __attribute__((ext_vector_type(8)));

#define NUM_EXPERTS 64
#define HDIM 2048
#define NTOK 16384   // B*S = 4*4096

// d_out layout (floats), reference tuple order:
#define OFF_DISPATCH 0
#define OFF_COMBINE  (NTOK * NUM_EXPERTS)
#define OFF_LOGITS   (2 * NTOK * NUM_EXPERTS)
#define OFF_AUX      (3 * NTOK * NUM_EXPERTS)
#define OFF_Z        (OFF_AUX + 1)
#define OFF_IDX      (OFF_AUX + 2)
#define OFF_ENT      (OFF_IDX + NTOK * 2)

// ---------------------------------------------------------------------------
// 0) zero the workspace accumulators (graph-capture safe)
// ---------------------------------------------------------------------------
__global__ void zero_ws(float* ws, int n) {
    int i = blockIdx.x * blockDim.x + threadIdx.x;
    if (i < n) ws[i] = 0.0f;
}

// ---------------------------------------------------------------------------
// 1) router GEMM: logits[t][e] = sum_h hidden[t][h] * gate[e][h]
//    fp32 WMMA 16x16x4; one wave handles a 16(M) x 64(N) tile, K = 2048.
//    A 16x4 layout: lanes 0-15 row M=lane (K=k,k+1), lanes 16-31 same rows
//    (K=k+2,k+3) -> per-lane float2 at hidden[row*H + khalf + k].
//    B 4x16 layout mirrors it on gate rows (experts).
// ---------------------------------------------------------------------------
__global__ __launch_bounds__(128) void router_gemm_wmma(
    const float* __restrict__ hidden, const float* __restrict__ gate,
    float* __restrict__ logits)
{
    const int lane  = threadIdx.x & 31;
    const int wave  = threadIdx.x >> 5;
    const int tile  = blockIdx.x * 4 + wave;    // 1024 tiles of 16 tokens
    const int rowBase = tile * 16;
    const int mrow  = lane & 15;
    const int khalf = (lane >> 4) << 1;         // 0 or 2

    const float* __restrict__ aptr  = hidden + (size_t)(rowBase + mrow) * HDIM + khalf;
    const float* __restrict__ bptr0 = gate + (size_t)( 0 + mrow) * HDIM + khalf;
    const float* __restrict__ bptr1 = gate + (size_t)(16 + mrow) * HDIM + khalf;
    const float* __restrict__ bptr2 = gate + (size_t)(32 + mrow) * HDIM + khalf;
    const float* __restrict__ bptr3 = gate + (size_t)(48 + mrow) * HDIM + khalf;

    v8f acc0 = {}, acc1 = {}, acc2 = {}, acc3 = {};

    #pragma unroll 4
    for (int k = 0; k < HDIM; k += 4) {
        v2f a  = *(const v2f*)(aptr  + k);
        v2f b0 = *(const v2f*)(bptr0 + k);
        v2f b1 = *(const v2f*)(bptr1 + k);
        v2f b2 = *(const v2f*)(bptr2 + k);
        v2f b3 = *(const v2f*)(bptr3 + k);
        acc0 = __builtin_amdgcn_wmma_f32_16x16x4_f32(false, a, false, b0, (short)0, acc0, false, false);
        acc1 = __builtin_amdgcn_wmma_f32_16x16x4_f32(false, a, false, b1, (short)0, acc1, false, false);
        acc2 = __builtin_amdgcn_wmma_f32_16x16x4_f32(false, a, false, b2, (short)0, acc2, false, false);
        acc3 = __builtin_amdgcn_wmma_f32_16x16x4_f32(false, a, false, b3, (short)0, acc3, false, false);
    }

    // C/D layout: VGPR v: lanes 0-15 -> M=v, lanes 16-31 -> M=v+8; N=lane&15
    const int colBase = lane & 15;
    const int rOff    = (lane >> 4) * 8;
    float* __restrict__ outp =
        logits + (size_t)(rowBase + rOff) * NUM_EXPERTS + colBase;
    #pragma unroll
    for (int v = 0; v < 8; ++v) {
        outp[(size_t)v * NUM_EXPERTS +  0] = acc0[v];
        outp[(size_t)v * NUM_EXPERTS + 16] = acc1[v];
        outp[(size_t)v * NUM_EXPERTS + 32] = acc2[v];
        outp[(size_t)v * NUM_EXPERTS + 48] = acc3[v];
    }
}

// ---------------------------------------------------------------------------
// 2) per-token routing: one wave32 per token, lane covers experts {l, l+32}
// ---------------------------------------------------------------------------
__device__ __forceinline__ bool vi_gt(float v1, int i1, float v2, int i2) {
    // JAX top_k tie-break: lower index first
    return (v1 > v2) || ((v1 == v2) && (i1 < i2));
}

__global__ __launch_bounds__(256) void router_topk(
    const float* __restrict__ logits, float* __restrict__ out,
    float* __restrict__ ws)
{
    __shared__ float s_avgp[64];
    __shared__ float s_tok[64];
    __shared__ float s_z, s_ent;

    const int tid = threadIdx.x;
    if (tid < 64) { s_avgp[tid] = 0.0f; s_tok[tid] = 0.0f; }
    if (tid == 0) { s_z = 0.0f; s_ent = 0.0f; }
    __syncthreads();

    const int lane  = tid & 31;
    const int token = blockIdx.x * 8 + (tid >> 5);
    const float* __restrict__ lg = logits + (size_t)token * NUM_EXPERTS;

    const float l0 = lg[lane];
    const float l1 = lg[lane + 32];

    // --- logsumexp over 64 logits ---
    float m = fmaxf(l0, l1);
    #pragma unroll
    for (int off = 16; off >= 1; off >>= 1) m = fmaxf(m, __shfl_xor(m, off, 32));
    float ssum = __expf(l0 - m) + __expf(l1 - m);
    #pragma unroll
    for (int off = 16; off >= 1; off >>= 1) ssum += __shfl_xor(ssum, off, 32);
    const float lse = m + __logf(ssum);

    const float p0 = __expf(l0 - lse);
    const float p1 = __expf(l1 - lse);

    // entropy_t = lse - sum(p * l)
    float pl = p0 * l0 + p1 * l1;
    #pragma unroll
    for (int off = 16; off >= 1; off >>= 1) pl += __shfl_xor(pl, off, 32);
    const float ent = lse - pl;

    // --- top-2 (value,index) butterfly merge ---
    float av, bv; int ai, bi;
    if (vi_gt(l0, lane, l1, lane + 32)) { av = l0; ai = lane;      bv = l1; bi = lane + 32; }
    else                                { av = l1; ai = lane + 32; bv = l0; bi = lane; }
    #pragma unroll
    for (int off = 16; off >= 1; off >>= 1) {
        float av2 = __shfl_xor(av, off, 32);
        int   ai2 = __shfl_xor(ai, off, 32);
        float bv2 = __shfl_xor(bv, off, 32);
        int   bi2 = __shfl_xor(bi, off, 32);
        if (vi_gt(av, ai, av2, ai2)) {
            if (vi_gt(av2, ai2, bv, bi)) { bv = av2; bi = ai2; }
        } else {
            if (vi_gt(av, ai, bv2, bi2)) { bv = av; bi = ai; }
            else                         { bv = bv2; bi = bi2; }
            av = av2; ai = ai2;
        }
    }
    // softmax over the 2 selected logits (av >= bv)
    const float r     = __expf(bv - av);
    const float w0    = 1.0f / (1.0f + r);
    const float w1    = r / (1.0f + r);

    // --- per-expert outputs for this lane's two experts ---
    const int eA = lane, eB = lane + 32;
    const float dmA = (eA == ai ? 1.0f : 0.0f) + (eA == bi ? 1.0f : 0.0f);
    const float dmB = (eB == ai ? 1.0f : 0.0f) + (eB == bi ? 1.0f : 0.0f);
    const float cwA = (eA == ai ? w0 : 0.0f) + (eA == bi ? w1 : 0.0f);
    const float cwB = (eB == ai ? w0 : 0.0f) + (eB == bi ? w1 : 0.0f);

    const size_t tb = (size_t)token * NUM_EXPERTS;
    out[OFF_DISPATCH + tb + eA] = dmA;
    out[OFF_DISPATCH + tb + eB] = dmB;
    out[OFF_COMBINE  + tb + eA] = cwA;
    out[OFF_COMBINE  + tb + eB] = cwB;
    if (lane == 0) {
        out[OFF_IDX + (size_t)token * 2 + 0] = (float)ai;
        out[OFF_IDX + (size_t)token * 2 + 1] = (float)bi;
    }

    // --- block-local accumulation, then one global atomic per expert ---
    atomicAdd(&s_avgp[eA], p0);
    atomicAdd(&s_avgp[eB], p1);
    if (dmA != 0.0f) atomicAdd(&s_tok[eA], dmA);
    if (dmB != 0.0f) atomicAdd(&s_tok[eB], dmB);
    if (lane == 0) {
        atomicAdd(&s_z,   lse * lse);
        atomicAdd(&s_ent, ent);
    }
    __syncthreads();
    if (tid < 64) {
        atomicAdd(&ws[tid],      s_avgp[tid]);
        atomicAdd(&ws[64 + tid], s_tok[tid]);
    }
    if (tid == 0) {
        atomicAdd(&ws[128], s_z);
        atomicAdd(&ws[129], s_ent);
    }
}

// ---------------------------------------------------------------------------
// 3) finalize the scalar losses (single wave)
// ---------------------------------------------------------------------------
__global__ __launch_bounds__(32) void router_finalize(
    const float* __restrict__ ws, float* __restrict__ out)
{
    const int lane = threadIdx.x;
    const float T   = (float)NTOK;
    const float tot = 2.0f * T + 1e-6f;

    float partial = 0.0f;
    #pragma unroll
    for (int e = lane; e < 64; e += 32)
        partial += (ws[e] / T) * (ws[64 + e] / tot);
    #pragma unroll
    for (int off = 16; off >= 1; off >>= 1)
        partial += __shfl_xor(partial, off, 32);

    if (lane == 0) {
        out[OFF_AUX] = partial * (float)NUM_EXPERTS * 0.01f;  // aux loss
        out[OFF_Z]   = (ws[128] / T) * 0.001f;                // z loss
        out[OFF_ENT] = ws[129] / T;                           // routing entropy
    }
}

// ---------------------------------------------------------------------------
extern "C" void kernel_launch(void* const* d_in, const int* in_sizes, int n_in,
                              void* d_out, int out_size, void* d_ws, size_t ws_size,
                              hipStream_t stream) {
    const float* hidden = (const float*)d_in[0];  // [4,4096,2048] f32
    const float* gate   = (const float*)d_in[1];  // [64,2048] f32
    float* out = (float*)d_out;
    float* ws  = (float*)d_ws;

    zero_ws<<<1, 256, 0, stream>>>(ws, 130);
    router_gemm_wmma<<<256, 128, 0, stream>>>(hidden, gate, out + OFF_LOGITS);
    router_topk<<<NTOK / 8, 256, 0, stream>>>(out + OFF_LOGITS, out, ws);
    router_finalize<<<1, 32, 0, stream>>>(ws, out);
}